// ComplexAudioLayerScene_23545010717463
// MI455X (gfx1250) — compile-verified
//
#include <hip/hip_runtime.h>
#include <math.h>

#define K_OSC    64
#define T_FRAMES 1024
#define F_BINS   1025
#define H_HARM   16
#define N_CTRL   257
#define F_MIN_BIN 3.7151927437641724f   // 40*2048/22050
#define F_MAX_BIN 1024.0f
#define SIG_MIN  0.5f
#define SIG_MAX  60.0f
#define TT_FLOOR 0.1f                   // 1 - T_MAX
#define NFB 5                           // f-blocks of 256 covering 1025
#define NTB 8                           // t-chunks of 128

typedef float v8f __attribute__((ext_vector_type(8)));
typedef float v2f __attribute__((ext_vector_type(2)));

// Workspace layout (float units). Total ~1.85 MB.
enum : int {
  OFF_PATH   = 0,                       // [K*T] raw interp of path_ctrl
  OFF_ALPHAI = 65536,                   // [K*T] raw interp of alpha_ctrl
  OFF_PHASE  = 131072,                  // [K*T] raw interp of phase_ctrl
  OFF_PACK   = 196608,                  // [K*T] float4 {freq, alpha, cos, sin}
  OFF_HARM   = 458752,                  // [K*16] softmax
  OFF_INV0   = 459776,                  // [K] 1/(2*sigma^2)
  OFF_INV1   = 459840,                  // [K] 1/(2*(0.7*sigma)^2)
  OFF_CUT    = 459904,                  // [K] distance at which exp underflows to 0
  OFF_PART   = 459968,                  // [K*NTB*NFB] salience partials
  OFF_SAL    = 462528,                  // [K]
  OFF_ORDER  = 462592                   // [K] int
};

// ---------------------------------------------------------------- prep
__global__ void prep_kernel(const float* __restrict__ log_sigma_f,
                            const float* __restrict__ logits,
                            float* __restrict__ ws) {
  int k = threadIdx.x;
  if (k >= K_OSC) return;
  float l[H_HARM];
  float m = -1e30f;
  for (int h = 0; h < H_HARM; ++h) { l[h] = logits[k * H_HARM + h]; m = fmaxf(m, l[h]); }
  float s = 0.f;
  for (int h = 0; h < H_HARM; ++h) { l[h] = __expf(l[h] - m); s += l[h]; }
  float inv_s = 1.0f / s;
  for (int h = 0; h < H_HARM; ++h) ws[OFF_HARM + k * H_HARM + h] = l[h] * inv_s;
  float sigma = fminf(fmaxf(__expf(log_sigma_f[k]), SIG_MIN), SIG_MAX);
  float s07 = 0.7f * sigma;
  ws[OFF_INV0 + k] = 1.0f / (2.0f * sigma * sigma);
  ws[OFF_INV1 + k] = 1.0f / (2.0f * s07 * s07);
  // exp(-d^2 * inv1) == 0 in fp32 once d^2*inv1 > ~104  (below min denormal)
  ws[OFF_CUT + k] = sqrtf(104.0f * 2.0f) * s07;   // = sqrt(104/inv1)
}

// --------------------------------------------------- interp via f32 WMMA
// out[64,1024] = ctrl[64,257] x W[257,1024], W = banded lerp weights.
// 16x16 output tiles; only the 2-3 nonzero K-chunks of 4 are multiplied.
// A-index is clamped (not guarded): any out-of-band control point gets
// weight 0 in B, so clamped (finite) A data contributes exactly 0.
__global__ __launch_bounds__(32) void interp_wmma(const float* __restrict__ pc,
                                                  const float* __restrict__ ac,
                                                  const float* __restrict__ phc,
                                                  float* __restrict__ ws) {
  int bid  = blockIdx.x;
  int sel  = bid >> 8;        // 0..2 : which control array
  int tile = bid & 255;
  int kb   = tile >> 6;       // 0..3  row tile
  int tb   = tile & 63;       // 0..63 col tile
  const float* src = (sel == 0) ? pc : ((sel == 1) ? ac : phc);
  float* dst = ws + ((sel == 0) ? OFF_PATH : (sel == 1) ? OFF_ALPHAI : OFF_PHASE);

  int lane = threadIdx.x;
  int half = lane >> 4;       // 0|1
  int m    = lane & 15;

  const float step = 256.0f / 1023.0f;            // (N_CTRL-1)/(T-1)
  int t0 = tb * 16;
  int lo0  = min((int)floorf((float)t0 * step),        N_CTRL - 2);
  int lo15 = min((int)floorf((float)(t0 + 15) * step), N_CTRL - 2);
  int c0 = lo0 >> 2, c1 = (lo15 + 1) >> 2;        // wave-uniform chunk range

  int   tcol = t0 + m;
  float pos  = (float)tcol * step;
  int   lo   = min((int)floorf(pos), N_CTRL - 2);
  float frac = pos - (float)lo;

  const float* rowp = src + (kb * 16 + m) * N_CTRL;
  v8f acc = {};
  for (int c = c0; c <= c1; ++c) {
    int n0  = c * 4;
    int na0 = n0 + 2 * half;
    int na1 = na0 + 1;
    v2f A, B;
    A.x = rowp[min(na0, N_CTRL - 1)];             // unconditional, clamped
    A.y = rowp[min(na1, N_CTRL - 1)];
    B.x = (na0 == lo) ? (1.f - frac) : ((na0 == lo + 1) ? frac : 0.f);
    B.y = (na1 == lo) ? (1.f - frac) : ((na1 == lo + 1) ? frac : 0.f);
    acc = __builtin_amdgcn_wmma_f32_16x16x4_f32(false, A, false, B,
                                                (short)0, acc, false, false);
  }
  #pragma unroll
  for (int j = 0; j < 8; ++j) {
    int row = kb * 16 + j + 8 * half;             // D layout: VGPR j -> M=j / j+8
    dst[row * T_FRAMES + tcol] = acc[j];
  }
}

// ---------------------------------------------------------------- post
__global__ __launch_bounds__(256) void post_kernel(const float* __restrict__ mu,
                                                   float* __restrict__ ws) {
  int i = blockIdx.x * 256 + threadIdx.x;
  if (i >= K_OSC * T_FRAMES) return;
  int k = i >> 10;                                // /T_FRAMES
  float freq = fminf(fmaxf(mu[k] + ws[OFF_PATH + i], F_MIN_BIN), F_MAX_BIN);
  float a = 1.0f / (1.0f + __expf(-ws[OFF_ALPHAI + i]));
  float s, c;
  sincosf(ws[OFF_PHASE + i], &s, &c);
  float4 p;  p.x = freq;  p.y = a;  p.z = c;  p.w = s;
  ((float4*)(ws + OFF_PACK))[i] = p;              // one b128 store
}

// --------------------------------------------- Gaussian comb magnitude
__device__ __forceinline__ float gmag(float fv, float freq, const float* hh,
                                      float i0, float i1, float cut) {
  float d0  = fv - freq;
  float mag = hh[0] * __expf(-d0 * d0 * i0);
  float c   = freq;
  #pragma unroll 5
  for (int h = 1; h < H_HARM; ++h) {
    c += freq;                                    // center = freq*(h+1)
    float d = fv - c;
    if (d < -cut) break;                          // all later harmonics underflow
    mag += hh[h] * __expf(-d * d * i1);
  }
  return mag;
}

// ------------------------------------------------------------- salience
__global__ __launch_bounds__(256) void salience_partial(float* __restrict__ ws) {
  int fb = blockIdx.x, k = blockIdx.y, tb = blockIdx.z;
  int f  = fb * 256 + threadIdx.x;
  float hh[H_HARM];
  #pragma unroll
  for (int h = 0; h < H_HARM; ++h) hh[h] = ws[OFF_HARM + k * H_HARM + h];
  float i0 = ws[OFF_INV0 + k], i1 = ws[OFF_INV1 + k], cut = ws[OFF_CUT + k];
  const float4* pack = (const float4*)(ws + OFF_PACK);
  float acc = 0.f;
  if (f < F_BINS) {
    float fv = (float)f;
    for (int t = tb * 128; t < tb * 128 + 128; ++t) {
      float4 p = pack[k * T_FRAMES + t];          // {freq, alpha, cos, sin}
      float mag  = gmag(fv, p.x, hh, i0, i1, cut);
      float r = mag * p.z, im = mag * p.w;
      acc += p.y * sqrtf(r * r + im * im + 1e-12f);
    }
  }
  __shared__ float red[256];
  red[threadIdx.x] = acc;
  __syncthreads();
  for (int s = 128; s > 0; s >>= 1) {
    if (threadIdx.x < s) red[threadIdx.x] += red[threadIdx.x + s];
    __syncthreads();
  }
  if (threadIdx.x == 0) ws[OFF_PART + (k * NTB + tb) * NFB + fb] = red[0];
}

__global__ void salience_reduce(float* __restrict__ ws) {
  int k = threadIdx.x;
  if (k >= K_OSC) return;
  float s = 0.f;
  for (int i = 0; i < NTB * NFB; ++i) s += ws[OFF_PART + k * NTB * NFB + i];
  ws[OFF_SAL + k] = s;
}

// Stable rank == jnp.argsort(-salience)
__global__ void order_kernel(float* __restrict__ ws) {
  __shared__ float s[K_OSC];
  int k = threadIdx.x;
  if (k < K_OSC) s[k] = ws[OFF_SAL + k];
  __syncthreads();
  if (k >= K_OSC) return;
  float mine = s[k];
  int rank = 0;
  for (int j = 0; j < K_OSC; ++j)
    rank += (s[j] > mine) || (s[j] == mine && j < k);
  ((int*)(ws + OFF_ORDER))[rank] = k;
}

// ------------------------------------------------- transmittance scan
__global__ __launch_bounds__(256) void scan_kernel(const float* __restrict__ ws,
                                                   float* __restrict__ out) {
  __shared__ float sh_harm[K_OSC * H_HARM];
  __shared__ float sh_i0[K_OSC], sh_i1[K_OSC], sh_cut[K_OSC];
  __shared__ int   sh_ord[K_OSC];
  int tx = threadIdx.x;
  for (int i = tx; i < K_OSC * H_HARM; i += 256) sh_harm[i] = ws[OFF_HARM + i];
  if (tx < K_OSC) {
    sh_i0[tx]  = ws[OFF_INV0 + tx];
    sh_i1[tx]  = ws[OFF_INV1 + tx];
    sh_cut[tx] = ws[OFF_CUT + tx];
    sh_ord[tx] = ((const int*)(ws + OFF_ORDER))[tx];
  }
  __syncthreads();

  int g = blockIdx.x * 256 + tx;
  if (g >= T_FRAMES * F_BINS) return;
  int t = g / F_BINS;
  int f = g - t * F_BINS;
  float fv = (float)f;
  const float4* pack = (const float4*)(ws + OFF_PACK);

  float outr = 0.f, outi = 0.f, tt = 1.0f;
  for (int kk = 0; kk < K_OSC; ++kk) {
    int k = sh_ord[kk];
    float4 p = pack[k * T_FRAMES + t];            // {freq, alpha, cos, sin}
    float mag  = gmag(fv, p.x, &sh_harm[k * H_HARM], sh_i0[k], sh_i1[k], sh_cut[k]);
    float r = mag * p.z, im = mag * p.w;
    float magc = sqrtf(r * r + im * im + 1e-12f);
    float aeff = fminf(fmaxf(p.y * magc, 0.f), 1.f);
    float w = tt * p.y;
    outr = fmaf(w, r, outr);
    outi = fmaf(w, im, outi);
    tt = fmaxf(tt * (1.0f - aeff), TT_FLOOR);
  }
  out[2 * g]     = outr;   // complex64 interleaved
  out[2 * g + 1] = outi;
}

// ---------------------------------------------------------------- host
extern "C" void kernel_launch(void* const* d_in, const int* in_sizes, int n_in,
                              void* d_out, int out_size, void* d_ws, size_t ws_size,
                              hipStream_t stream) {
  const float* mu_f        = (const float*)d_in[0];
  const float* log_sigma_f = (const float*)d_in[1];
  const float* path_ctrl   = (const float*)d_in[2];
  const float* alpha_ctrl  = (const float*)d_in[3];
  const float* harm_logits = (const float*)d_in[4];
  const float* phase_ctrl  = (const float*)d_in[5];
  float* ws  = (float*)d_ws;
  float* out = (float*)d_out;

  prep_kernel<<<1, 64, 0, stream>>>(log_sigma_f, harm_logits, ws);
  interp_wmma<<<768, 32, 0, stream>>>(path_ctrl, alpha_ctrl, phase_ctrl, ws);
  post_kernel<<<(K_OSC * T_FRAMES + 255) / 256, 256, 0, stream>>>(mu_f, ws);
  salience_partial<<<dim3(NFB, K_OSC, NTB), 256, 0, stream>>>(ws);
  salience_reduce<<<1, 64, 0, stream>>>(ws);
  order_kernel<<<1, 64, 0, stream>>>(ws);
  scan_kernel<<<(T_FRAMES * F_BINS + 255) / 256, 256, 0, stream>>>(ws, out);
}